// MeshPoolTrans_3633542332722
// MI455X (gfx1250) — compile-verified
//
#include <hip/hip_runtime.h>

// Problem constants (fixed by the reference harness).
#define MP_N   10242   // pooled (output) vertices
#define M_N    40962   // fine vertices
#define F_N    128     // features
#define B_N    16      // batch
#define POOL_BLOCK 256
#define SEG_CAP 512    // max nonzeros per row handled via sorted/staged fast path

// ---------------------------------------------------------------------------
// CSR build: counts -> exclusive scan -> stable-ish scatter of permutation k.
// ---------------------------------------------------------------------------
__global__ void zero_counts_kernel(int* __restrict__ counts, int n) {
  int i = blockIdx.x * blockDim.x + threadIdx.x;
  if (i < n) counts[i] = 0;
}

__global__ void count_kernel(const int* __restrict__ rows,
                             int* __restrict__ counts, int nnz) {
  int k = blockIdx.x * blockDim.x + threadIdx.x;
  if (k < nnz) atomicAdd(&counts[rows[k]], 1);
}

// Single-block exclusive scan over MP_N counts (11 chunks of 1024).
__global__ __launch_bounds__(1024)
void scan_kernel(const int* __restrict__ counts, int* __restrict__ offsets,
                 int* __restrict__ cursor, int mp) {
  __shared__ int sdata[1024];
  __shared__ int carry;
  const int tid = threadIdx.x;
  if (tid == 0) carry = 0;
  __syncthreads();
  for (int base = 0; base < mp; base += 1024) {
    int i = base + tid;
    int v = (i < mp) ? counts[i] : 0;
    sdata[tid] = v;
    __syncthreads();
    for (int off = 1; off < 1024; off <<= 1) {
      int t = (tid >= off) ? sdata[tid - off] : 0;
      __syncthreads();
      sdata[tid] += t;
      __syncthreads();
    }
    int excl  = sdata[tid] - v;   // exclusive within chunk
    int c0    = carry;            // everyone reads old carry
    int total = sdata[1023];      // chunk total
    __syncthreads();
    if (tid == 0) carry = c0 + total;
    __syncthreads();
    if (i < mp) { int o = c0 + excl; offsets[i] = o; cursor[i] = o; }
  }
  if (tid == 0) offsets[mp] = carry;
}

__global__ void scatter_kernel(const int* __restrict__ rows,
                               int* __restrict__ cursor,
                               int* __restrict__ csr_k, int nnz) {
  int k = blockIdx.x * blockDim.x + threadIdx.x;
  if (k < nnz) {
    int pos = atomicAdd(&cursor[rows[k]], 1);  // order fixed up by sort later
    csr_k[pos] = k;
  }
}

// ---------------------------------------------------------------------------
// Pool kernel: one block per output row p. Async global->LDS double-buffered
// gather of x[b, col, :] (CDNA5 ASYNCcnt path), register accumulation,
// single coalesced b128 store of the output row.
// ---------------------------------------------------------------------------
__device__ __forceinline__ unsigned lds_addr32(const void* p) {
  // Generic pointers to __shared__ carry the LDS offset in their low 32 bits.
  return (unsigned)(unsigned long long)p;
}

__global__ __launch_bounds__(POOL_BLOCK)
void pool_kernel(const float* __restrict__ x, const int* __restrict__ cols,
                 const float* __restrict__ vals, const int* __restrict__ offsets,
                 const int* __restrict__ csr_k, float* __restrict__ out) {
  __shared__ int   s_k[SEG_CAP];
  __shared__ int   s_col[SEG_CAP];
  __shared__ float s_val[SEG_CAP];
  __shared__ float buf[2][B_N * F_N / 4 * 4];  // 2 x 2048 floats = 16 KB

  const int p     = blockIdx.x;
  const int tid   = threadIdx.x;
  const int start = offsets[p];
  const int L     = offsets[p + 1] - start;
  const int n     = (L < SEG_CAP) ? L : SEG_CAP;

  // Deterministic order: sort the row's original nonzero indices k ascending.
  if (tid == 0) {
    for (int i = 0; i < n; ++i) {
      int kk = csr_k[start + i];
      int j = i;
      while (j > 0 && s_k[j - 1] > kk) { s_k[j] = s_k[j - 1]; --j; }
      s_k[j] = kk;
    }
  }
  __syncthreads();
  for (int i = tid; i < n; i += POOL_BLOCK) {
    int kk = s_k[i];
    s_col[i] = cols[kk];
    s_val[i] = vals[kk];
  }
  __syncthreads();

  // Each thread owns two float4 output slots across the (b, f) plane.
  const int e0 = tid, e1 = tid + POOL_BLOCK;        // slots in [0, 512)
  const int b0 = e0 >> 5, f0 = (e0 & 31) << 2;      // b in [0,16), f multiple of 4
  const int b1 = e1 >> 5, f1 = (e1 & 31) << 2;
  float4 acc0 = make_float4(0.f, 0.f, 0.f, 0.f);
  float4 acc1 = make_float4(0.f, 0.f, 0.f, 0.f);

  auto issue = [&](int j) {
    const int bsel = j & 1;
    const int col  = s_col[j];
    const float* g0 = x + ((size_t)b0 * M_N + col) * F_N + f0;
    const float* g1 = x + ((size_t)b1 * M_N + col) * F_N + f1;
    const unsigned l0 = lds_addr32(&buf[bsel][e0 * 4]);
    const unsigned l1 = lds_addr32(&buf[bsel][e1 * 4]);
    // Ensure prior ds_loads from this buffer parity have completed before the
    // async engine (which is unordered vs DS) overwrites it.
    asm volatile("s_wait_dscnt 0" ::: "memory");
    asm volatile("global_load_async_to_lds_b128 %0, %1, off"
                 :: "v"(l0), "v"(g0) : "memory");
    asm volatile("global_load_async_to_lds_b128 %0, %1, off"
                 :: "v"(l1), "v"(g1) : "memory");
  };

  if (n > 0) issue(0);
  for (int j = 0; j < n; ++j) {
    if (j + 1 < n) {
      issue(j + 1);                                  // prefetch next nonzero
      asm volatile("s_wait_asynccnt 2" ::: "memory"); // j's 2 copies done (in-order)
    } else {
      asm volatile("s_wait_asynccnt 0" ::: "memory");
    }
    const float v  = s_val[j];
    const int bsel = j & 1;
    const float4 d0 = *(const float4*)&buf[bsel][e0 * 4];
    const float4 d1 = *(const float4*)&buf[bsel][e1 * 4];
    acc0.x += v * d0.x; acc0.y += v * d0.y; acc0.z += v * d0.z; acc0.w += v * d0.w;
    acc1.x += v * d1.x; acc1.y += v * d1.y; acc1.z += v * d1.z; acc1.w += v * d1.w;
  }

  // Pathological overflow tail (L > SEG_CAP): plain synchronous gather.
  for (int j = n; j < L; ++j) {
    int kk = csr_k[start + j];
    int col = cols[kk];
    float v = vals[kk];
    const float4 d0 = *(const float4*)(x + ((size_t)b0 * M_N + col) * F_N + f0);
    const float4 d1 = *(const float4*)(x + ((size_t)b1 * M_N + col) * F_N + f1);
    acc0.x += v * d0.x; acc0.y += v * d0.y; acc0.z += v * d0.z; acc0.w += v * d0.w;
    acc1.x += v * d1.x; acc1.y += v * d1.y; acc1.z += v * d1.z; acc1.w += v * d1.w;
  }

  *(float4*)(out + ((size_t)b0 * MP_N + p) * F_N + f0) = acc0;
  *(float4*)(out + ((size_t)b1 * MP_N + p) * F_N + f1) = acc1;
}

// ---------------------------------------------------------------------------
extern "C" void kernel_launch(void* const* d_in, const int* in_sizes, int n_in,
                              void* d_out, int out_size, void* d_ws, size_t ws_size,
                              hipStream_t stream) {
  const float* x    = (const float*)d_in[0];
  const int*   rows = (const int*)d_in[1];
  const int*   cols = (const int*)d_in[2];
  const float* vals = (const float*)d_in[3];
  float*       out  = (float*)d_out;
  const int    nnz  = in_sizes[1];

  // Workspace carve-up (all int32): counts[MP], offsets[MP+1], cursor[MP], csr_k[nnz]
  int* counts  = (int*)d_ws;
  int* offsets = counts + MP_N;
  int* cursor  = offsets + MP_N + 1;
  int* csr_k   = cursor + MP_N;

  zero_counts_kernel<<<(MP_N + 255) / 256, 256, 0, stream>>>(counts, MP_N);
  count_kernel<<<(nnz + 255) / 256, 256, 0, stream>>>(rows, counts, nnz);
  scan_kernel<<<1, 1024, 0, stream>>>(counts, offsets, cursor, MP_N);
  scatter_kernel<<<(nnz + 255) / 256, 256, 0, stream>>>(rows, cursor, csr_k, nnz);
  pool_kernel<<<MP_N, POOL_BLOCK, 0, stream>>>(x, cols, vals, offsets, csr_k, out);
}